// GaussianVoxel_83889301225807
// MI455X (gfx1250) — compile-verified
//
#include <hip/hip_runtime.h>
#include <hip/hip_bf16.h>
#include <stdint.h>

// ---------------------------------------------------------------------------
// GaussianVoxel: write small Gaussian patches into zero [B,J,zres,64,64] grids.
// Store-bandwidth bound: 316 MB of output -> ~13.6us floor @ 23.3 TB/s.
// Pass 1: streaming NT b128 zero-fill (4x unrolled grid-stride).
// Pass 2: per-(b,j) patch staged in LDS, scattered to global with the CDNA5
//         Tensor Data Mover (tensor_store_from_lds), one D# per level.
// ---------------------------------------------------------------------------

typedef float        v4f __attribute__((ext_vector_type(4)));
typedef unsigned int v4u __attribute__((ext_vector_type(4)));
typedef int          v4i __attribute__((ext_vector_type(4)));
typedef int          v8i __attribute__((ext_vector_type(8)));

#define PADC   6          // 3*HC
#define PATCHC 13         // 2*PAD+1

// level base offsets in floats (B*J = 272, plane = 64*64 = 4096)
#define L0BASE 0
#define L1BASE 1114112    // + 272*1*4096
#define L2BASE 3342336    // + 272*2*4096
#define L3BASE 7798784    // + 272*4*4096

// ---------------------------------------------------------------------------
// Pass 1: zero-fill with non-temporal 128-bit stores (grid-stride, 4x unroll).
// ---------------------------------------------------------------------------
__global__ void GV_fill_zero(v4f* __restrict__ out4, int n4, float* __restrict__ out,
                             int n_total) {
  const v4f z = {0.f, 0.f, 0.f, 0.f};
  const int stride = gridDim.x * blockDim.x;
  int i = blockIdx.x * blockDim.x + threadIdx.x;
  for (; i + 3 * stride < n4; i += 4 * stride) {
    __builtin_nontemporal_store(z, out4 + i);
    __builtin_nontemporal_store(z, out4 + i + stride);
    __builtin_nontemporal_store(z, out4 + i + 2 * stride);
    __builtin_nontemporal_store(z, out4 + i + 3 * stride);
  }
  for (; i < n4; i += stride)
    __builtin_nontemporal_store(z, out4 + i);
  // scalar tail (out_size is 4-divisible here, but stay robust)
  if (blockIdx.x == 0 && threadIdx.x == 0)
    for (int r = n4 * 4; r < n_total; ++r) out[r] = 0.f;
}

// ---------------------------------------------------------------------------
// TDM: store a dense [tdz, tdy, tdx] float tile from LDS into a strided
// [*, 64, 64] global tensor region starting at gptr.
// D# layout per CDNA5 ISA.
// ---------------------------------------------------------------------------
__device__ __forceinline__ unsigned gv_lds_byte_off(const void* p) {
  // flat shared pointer: addr[31:0] == wave-relative LDS byte offset
  return (unsigned)(uintptr_t)p;
}

__device__ __forceinline__ void gv_tdm_store_tile(unsigned lds_off, const float* gptr,
                                                  int tdx, int tdy, int tdz) {
  const unsigned long long ga = (unsigned long long)(uintptr_t)gptr;
  v4u d0;
  d0[0] = 1u;                                              // count=1 (valid), user mode
  d0[1] = lds_off;                                         // lds_addr (bytes)
  d0[2] = (unsigned)ga;                                    // global_addr[31:0]
  d0[3] = ((unsigned)(ga >> 32) & 0x01FFFFFFu)             // global_addr[56:32]
          | 0x80000000u;                                   // type=2 ("image")
  v8i d1;
  d1[0] = 0x00020000;                                      // data_size=2 (4 bytes)
  d1[1] = (tdx & 0xFFFF) << 16;                            // tensor_dim0[15:0]
  d1[2] = (tdy & 0xFFFF) << 16;                            // tdim0 hi=0 | tensor_dim1 lo
  d1[3] = (tdx & 0xFFFF) << 16;                            // tdim1 hi=0 | tile_dim0
  d1[4] = (tdy & 0xFFFF) | ((tdz & 0xFFFF) << 16);         // tile_dim1 | tile_dim2
  d1[5] = 64;                                              // tensor_dim0_stride lo32 (row)
  d1[6] = (int)(4096u << 16);                              // stride0 hi=0 | stride1 lo16
  d1[7] = 0;                                               // tensor_dim1_stride hi32
  v4i d2 = {tdz, 0, 0, 0};                                 // tensor_dim2; dim3/stride2 unused
  v4i d3 = {0, 0, 0, 0};
#if defined(__clang_major__) && __clang_major__ >= 23
  v8i d4 = {0, 0, 0, 0, 0, 0, 0, 0};
  __builtin_amdgcn_tensor_store_from_lds(d0, d1, d2, d3, d4, 0);
#else
  __builtin_amdgcn_tensor_store_from_lds(d0, d1, d2, d3, 0);
#endif
}

// ---------------------------------------------------------------------------
// Pass 2: one block per (b,j). Stage clipped Gaussian tiles into LDS, then
// wave 0 issues one TDM store per level. Patch iteration runs over the FULL
// 13x13(x13) index space so all div/mod use constant divisors, and the source
// value is just g[i] (patch coords == gauss coords by construction).
// ---------------------------------------------------------------------------
__global__ void GV_patch_scatter(const float* __restrict__ coords,
                                 const float* __restrict__ g0,
                                 const float* __restrict__ g1,
                                 const float* __restrict__ g2,
                                 const float* __restrict__ g3,
                                 float* __restrict__ out) {
  __shared__ float lds3[PATCHC * PATCHC * PATCHC];   // 2197 floats
  __shared__ float ldsA[PATCHC * PATCHC];            // level 0 plane
  __shared__ float ldsB[PATCHC * PATCHC];            // level 1 plane
  __shared__ float ldsC[PATCHC * PATCHC];            // level 2 plane

  const int bj = blockIdx.x;                         // 0 .. 271
  const float cx = coords[bj * 3 + 0];
  const float cy = coords[bj * 3 + 1];
  const float cz = coords[bj * 3 + 2];

  // force uniform (SGPR) scalars for descriptor construction
  const int xi  = __builtin_amdgcn_readfirstlane((int)cx);
  const int yi  = __builtin_amdgcn_readfirstlane((int)cy);
  const int zi0 = __builtin_amdgcn_readfirstlane((int)ceilf(cz * (1.0f / 64.0f)) - 1);
  const int zi1 = __builtin_amdgcn_readfirstlane((int)ceilf(cz * (2.0f / 64.0f)) - 1);
  const int zi2 = __builtin_amdgcn_readfirstlane((int)ceilf(cz * (4.0f / 64.0f)) - 1);
  const int zi3 = __builtin_amdgcn_readfirstlane((int)ceilf(cz) - 1);

  // clipped tile extents and start offsets inside the 13^k patch
  const int x0 = max(0, xi - PADC), tdx = min(64, xi + PADC + 1) - x0;
  const int y0 = max(0, yi - PADC), tdy = min(64, yi + PADC + 1) - y0;
  const int z0 = max(0, zi3 - PADC), tdz = min(64, zi3 + PADC + 1) - z0;
  const int sx = x0 - (xi - PADC);                   // first valid lx in [0,13)
  const int sy = y0 - (yi - PADC);
  const int sz = z0 - (zi3 - PADC);

  const int tid = threadIdx.x;

  // stage clipped 3D patch for level 3 (constant-divisor decomposition)
  for (int i = tid; i < PATCHC * PATCHC * PATCHC; i += (int)blockDim.x) {
    const int lz = i / (PATCHC * PATCHC);
    const int r  = i - lz * (PATCHC * PATCHC);
    const int ly = r / PATCHC;
    const int lx = r - ly * PATCHC;
    const int ux = lx - sx;                          // compact tile coords
    const int uy = ly - sy;
    const int uz = lz - sz;
    if ((unsigned)ux < (unsigned)tdx && (unsigned)uy < (unsigned)tdy &&
        (unsigned)uz < (unsigned)tdz)
      lds3[(uz * tdy + uy) * tdx + ux] = g3[i];
  }
  // stage clipped 2D planes for levels 0..2 (z_coeff = 1)
  {
    const float* gs[3] = {g0, g1, g2};
    float*       ls[3] = {ldsA, ldsB, ldsC};
#pragma unroll
    for (int l = 0; l < 3; ++l)
      for (int i = tid; i < PATCHC * PATCHC; i += (int)blockDim.x) {
        const int ly = i / PATCHC;
        const int lx = i - ly * PATCHC;
        const int ux = lx - sx;
        const int uy = ly - sy;
        if ((unsigned)ux < (unsigned)tdx && (unsigned)uy < (unsigned)tdy)
          ls[l][uy * tdx + ux] = gs[l][i];
      }
  }
  __syncthreads();   // LDS staging visible before TDM reads it

  if (tid < 32) {    // wave 0 only: issue the tensor DMA descriptors
    // level 3: [64,64,64] slice per (b,j)
    gv_tdm_store_tile(gv_lds_byte_off(lds3),
                      out + L3BASE + ((size_t)bj * 64 + (size_t)z0) * 4096
                          + (size_t)y0 * 64 + (size_t)x0,
                      tdx, tdy, tdz);
    // levels 0..2: single z-plane at zi (skip if the plane falls outside)
    if (zi0 >= 0 && zi0 < 1)
      gv_tdm_store_tile(gv_lds_byte_off(ldsA),
                        out + L0BASE + ((size_t)bj * 1 + (size_t)zi0) * 4096
                            + (size_t)y0 * 64 + (size_t)x0,
                        tdx, tdy, 1);
    if (zi1 >= 0 && zi1 < 2)
      gv_tdm_store_tile(gv_lds_byte_off(ldsB),
                        out + L1BASE + ((size_t)bj * 2 + (size_t)zi1) * 4096
                            + (size_t)y0 * 64 + (size_t)x0,
                        tdx, tdy, 1);
    if (zi2 >= 0 && zi2 < 4)
      gv_tdm_store_tile(gv_lds_byte_off(ldsC),
                        out + L2BASE + ((size_t)bj * 4 + (size_t)zi2) * 4096
                            + (size_t)y0 * 64 + (size_t)x0,
                        tdx, tdy, 1);
    __builtin_amdgcn_s_wait_tensorcnt(0);   // drain TDM before workgroup retires
  }
}

// ---------------------------------------------------------------------------
extern "C" void kernel_launch(void* const* d_in, const int* in_sizes, int n_in,
                              void* d_out, int out_size, void* d_ws, size_t ws_size,
                              hipStream_t stream) {
  const float* coords = (const float*)d_in[0];
  const float* g0     = (const float*)d_in[1];
  const float* g1     = (const float*)d_in[2];
  const float* g2     = (const float*)d_in[3];
  const float* g3     = (const float*)d_in[4];
  float* out = (float*)d_out;

  const int n4 = out_size >> 2;
  int blocks = (n4 + 255) / 256;
  if (blocks > 8192) blocks = 8192;
  GV_fill_zero<<<blocks, 256, 0, stream>>>((v4f*)out, n4, out, out_size);

  // 16*17 = 272 (b,j) pairs, one block each; runs after fill (stream order)
  GV_patch_scatter<<<272, 256, 0, stream>>>(coords, g0, g1, g2, g3, out);
}